// MambaLayers_2319282340139
// MI455X (gfx1250) — compile-verified
//
#include <hip/hip_runtime.h>

// ---------------------------------------------------------------------------
// Mamba 2-layer forward for MI455X (gfx1250, wave32).
//  * GEMMs via v_wmma_f32_16x16x4_f32, software-pipelined (next k-step's
//    fragments load while current k-step's 8 WMMAs issue), 32x64 tile/wave
//  * selective scan parallelized as a 3-pass chunked associative scan
//    (32 chunks of 128 -> 2048 concurrent waves instead of 64)
//  * B/C state vectors broadcast with wave32 __shfl (ds_bpermute)
//  * exp folded to native exp2 (v_exp_f32) with log2e pre-multiplied
// ---------------------------------------------------------------------------

typedef __attribute__((ext_vector_type(2))) float v2f;
typedef __attribute__((ext_vector_type(8))) float v8f;

#define BB      4
#define DMODEL  256
#define EDIM    512
#define NST     16
#define RNK     16
#define LSEQ    4096
#define MROWS   (BB * LSEQ)          // 16384
#define CL      128                  // scan chunk length
#define NCH     (LSEQ / CL)          // 32 chunks
#define LOG2E   1.44269504088896340736f

__device__ __forceinline__ float softplusf(float x) {
    return (x > 0.f) ? x + log1pf(expf(-x)) : log1pf(expf(x));
}
__device__ __forceinline__ float siluf(float x) {
    return x / (1.f + expf(-x));
}

// --------------------------- transposes ------------------------------------
// x (B, D, L) -> seq (B, L, D)
__global__ void transpose_in_k(const float* __restrict__ x, float* __restrict__ seq) {
    __shared__ float tile[32][33];
    const int b = blockIdx.z, l0 = blockIdx.x * 32, d0 = blockIdx.y * 32;
#pragma unroll
    for (int r = 0; r < 4; ++r) {
        int d = d0 + threadIdx.y + r * 8;
        tile[threadIdx.y + r * 8][threadIdx.x] =
            x[((size_t)(b * DMODEL + d)) * LSEQ + l0 + threadIdx.x];
    }
    __syncthreads();
#pragma unroll
    for (int r = 0; r < 4; ++r) {
        int l = l0 + threadIdx.y + r * 8;
        seq[((size_t)(b * LSEQ + l)) * DMODEL + d0 + threadIdx.x] =
            tile[threadIdx.x][threadIdx.y + r * 8];
    }
}

// seq (B, L, D) -> out (B, D, L)
__global__ void transpose_out_k(const float* __restrict__ seq, float* __restrict__ out) {
    __shared__ float tile[32][33];
    const int b = blockIdx.z, l0 = blockIdx.x * 32, d0 = blockIdx.y * 32;
#pragma unroll
    for (int r = 0; r < 4; ++r) {
        int l = l0 + threadIdx.y + r * 8;
        tile[threadIdx.y + r * 8][threadIdx.x] =
            seq[((size_t)(b * LSEQ + l)) * DMODEL + d0 + threadIdx.x];
    }
    __syncthreads();
#pragma unroll
    for (int r = 0; r < 4; ++r) {
        int d = d0 + threadIdx.y + r * 8;
        out[((size_t)(b * DMODEL + d)) * LSEQ + l0 + threadIdx.x] =
            tile[threadIdx.x][threadIdx.y + r * 8];
    }
}

// --------------------------- RMSNorm (1 wave / row) ------------------------
__global__ void rmsnorm_k(const float* __restrict__ seq, const float* __restrict__ nw,
                          float* __restrict__ out) {
    const int row  = (blockIdx.x * blockDim.x + threadIdx.x) >> 5;
    const int lane = threadIdx.x & 31;
    const float* p = seq + (size_t)row * DMODEL + lane * 8;
    float4 v0 = *(const float4*)(p);
    float4 v1 = *(const float4*)(p + 4);
    float ss = v0.x * v0.x + v0.y * v0.y + v0.z * v0.z + v0.w * v0.w +
               v1.x * v1.x + v1.y * v1.y + v1.z * v1.z + v1.w * v1.w;
#pragma unroll
    for (int m = 16; m >= 1; m >>= 1) ss += __shfl_xor(ss, m, 32);
    const float r = rsqrtf(ss * (1.f / DMODEL) + 1e-5f);
    const float* w = nw + lane * 8;
    float* q = out + (size_t)row * DMODEL + lane * 8;
    q[0] = v0.x * r * w[0]; q[1] = v0.y * r * w[1];
    q[2] = v0.z * r * w[2]; q[3] = v0.w * r * w[3];
    q[4] = v1.x * r * w[4]; q[5] = v1.y * r * w[5];
    q[6] = v1.z * r * w[6]; q[7] = v1.w * r * w[7];
}

// --------------------------- WMMA fp32 GEMM --------------------------------
// out[m,n] = sum_k A[m*ldA+k] * W[n*ldW+k]   (both operands K-major)
// Tile per wave: (16*MT) x (16*NT).  Software-pipelined fragment loads.
// EPI: 0 = plain store, 1 = softplus(acc + bias[n]), 2 = residual add (aux==out ok)
template <int MT, int NT, int EPI>
__global__ __launch_bounds__(128) void gemm_wmma_f32(
        const float* __restrict__ A, const float* __restrict__ W,
        const float* __restrict__ aux, float* __restrict__ out,
        int M, int K, int ldA, int ldW, int Nout) {
    const int wave = (blockIdx.x * blockDim.x + threadIdx.x) >> 5;
    const int lane = threadIdx.x & 31;
    const int hi   = lane >> 4;      // 0: K{0,1} / M 0..7, 1: K{2,3} / M 8..15
    const int lm   = lane & 15;
    const int nGroups = Nout / (16 * NT);
    const int mBase = (wave / nGroups) * (16 * MT);
    const int nBase = (wave % nGroups) * (16 * NT);
    if (mBase >= M) return;          // uniform per wave -> EXEC stays all-ones
    const int kb = hi * 2;

    const float* Ap[MT];
    const float* Wp[NT];
#pragma unroll
    for (int i = 0; i < MT; ++i) Ap[i] = A + (size_t)(mBase + i * 16 + lm) * ldA + kb;
#pragma unroll
    for (int j = 0; j < NT; ++j) Wp[j] = W + (size_t)(nBase + j * 16 + lm) * ldW + kb;

    v8f acc[MT][NT] = {};
    v2f a[MT], b[NT];
#pragma unroll
    for (int i = 0; i < MT; ++i) a[i] = *(const v2f*)(Ap[i]);
#pragma unroll
    for (int j = 0; j < NT; ++j) b[j] = *(const v2f*)(Wp[j]);

    for (int kk = 4; kk < K; kk += 4) {
        v2f an[MT], bn[NT];               // issue next k-step's loads first ...
#pragma unroll
        for (int i = 0; i < MT; ++i) an[i] = *(const v2f*)(Ap[i] + kk);
#pragma unroll
        for (int j = 0; j < NT; ++j) bn[j] = *(const v2f*)(Wp[j] + kk);
#pragma unroll
        for (int i = 0; i < MT; ++i)      // ... then overlap them with WMMAs
#pragma unroll
            for (int j = 0; j < NT; ++j)
                acc[i][j] = __builtin_amdgcn_wmma_f32_16x16x4_f32(
                    false, a[i], false, b[j], (short)0, acc[i][j], false, false);
#pragma unroll
        for (int i = 0; i < MT; ++i) a[i] = an[i];
#pragma unroll
        for (int j = 0; j < NT; ++j) b[j] = bn[j];
    }
#pragma unroll
    for (int i = 0; i < MT; ++i)
#pragma unroll
        for (int j = 0; j < NT; ++j)
            acc[i][j] = __builtin_amdgcn_wmma_f32_16x16x4_f32(
                false, a[i], false, b[j], (short)0, acc[i][j], false, false);

#pragma unroll
    for (int i = 0; i < MT; ++i) {
#pragma unroll
        for (int j = 0; j < NT; ++j) {
            const int n = nBase + j * 16 + lm;
#pragma unroll
            for (int v = 0; v < 8; ++v) {
                const int row = mBase + i * 16 + v + 8 * hi;
                const size_t o = (size_t)row * Nout + n;
                float val = acc[i][j][v];
                if (EPI == 1)      val = softplusf(val + aux[n]);
                else if (EPI == 2) val += aux[o];
                out[o] = val;
            }
        }
    }
}

// ---------------- depthwise causal conv (K=4) + bias + SiLU ----------------
__global__ void conv_silu_k(const float* __restrict__ xz, const float* __restrict__ cw,
                            const float* __restrict__ cb, float* __restrict__ xc) {
    const int gid = blockIdx.x * blockDim.x + threadIdx.x;   // (b, l, e) e-fastest
    const int e  = gid & (EDIM - 1);
    const int ml = gid >> 9;            // b*L + l
    const int l  = ml & (LSEQ - 1);
    float acc = cb[e];
#pragma unroll
    for (int j = 0; j < 4; ++j) {
        const int ll = l - 3 + j;
        if (ll >= 0)
            acc = fmaf(xz[((size_t)(ml - l + ll)) * 1024 + e], cw[e * 4 + j], acc);
    }
    xc[(size_t)gid] = siluf(acc);
}

// --------------------------- chunked selective scan ------------------------
// pass A: per chunk, per (b,e): local scan with h0=0 -> P[n]=prod(dA), S[n]=h_end
__global__ void scan_passA(const float* __restrict__ delta, const float* __restrict__ xc,
                           const float* __restrict__ dbc, const float* __restrict__ A_log,
                           float* __restrict__ P, float* __restrict__ S) {
    const int wave = (blockIdx.x * blockDim.x + threadIdx.x) >> 5;
    const int lane = threadIdx.x & 31;
    const int c   = wave % NCH;
    const int tmp = wave / NCH;
    const int eg  = tmp % (EDIM / 32);
    const int b   = tmp / (EDIM / 32);
    const int e   = eg * 32 + lane;

    float Av[NST], h[NST], Pr[NST];
#pragma unroll
    for (int n = 0; n < NST; ++n) {
        Av[n] = -expf(A_log[e * NST + n]) * LOG2E;   // exp(d*A) == exp2(d*Av)
        h[n] = 0.f; Pr[n] = 1.f;
    }
    const int l0 = c * CL;
    for (int t = 0; t < CL; ++t) {
        const size_t base = (size_t)(b * LSEQ + l0 + t);
        const float d  = delta[base * EDIM + e];
        const float xv = xc[base * EDIM + e];
        const float bc = dbc[base * 48 + 16 + lane];   // lanes 0-15: B, 16-31: C
        const float dx = d * xv;
#pragma unroll
        for (int n = 0; n < NST; ++n) {
            const float bn = __shfl(bc, n, 32);
            const float dA = exp2f(d * Av[n]);
            h[n] = fmaf(dA, h[n], dx * bn);
            Pr[n] *= dA;
        }
    }
#pragma unroll
    for (int n = 0; n < NST; ++n) {
        const size_t idx = ((size_t)((b * EDIM + e) * NST + n)) * NCH + c;
        P[idx] = Pr[n]; S[idx] = h[n];
    }
}

// pass B: tiny sequential scan over the 32 chunk boundaries per (b,e,n)
__global__ void scan_passB(const float* __restrict__ P, const float* __restrict__ S,
                           float* __restrict__ H) {
    const int gid = blockIdx.x * blockDim.x + threadIdx.x;   // < B*ED*NST
    const size_t base = (size_t)gid * NCH;
    float h = 0.f;
#pragma unroll
    for (int c = 0; c < NCH; ++c) {
        H[base + c] = h;
        h = fmaf(P[base + c], h, S[base + c]);
    }
}

// pass C: replay with correct h0, emit y, + D_skip*x, * silu(z); writes into xc
__global__ void scan_passC(const float* __restrict__ delta, float* __restrict__ xc,
                           const float* __restrict__ dbc, const float* __restrict__ A_log,
                           const float* __restrict__ Hinit, const float* __restrict__ Dskip,
                           const float* __restrict__ xz) {
    const int wave = (blockIdx.x * blockDim.x + threadIdx.x) >> 5;
    const int lane = threadIdx.x & 31;
    const int c   = wave % NCH;
    const int tmp = wave / NCH;
    const int eg  = tmp % (EDIM / 32);
    const int b   = tmp / (EDIM / 32);
    const int e   = eg * 32 + lane;

    float Av[NST], h[NST];
#pragma unroll
    for (int n = 0; n < NST; ++n) {
        Av[n] = -expf(A_log[e * NST + n]) * LOG2E;
        h[n]  = Hinit[((size_t)((b * EDIM + e) * NST + n)) * NCH + c];
    }
    const float dsk = Dskip[e];
    const int l0 = c * CL;
    for (int t = 0; t < CL; ++t) {
        const size_t base = (size_t)(b * LSEQ + l0 + t);
        const float d  = delta[base * EDIM + e];
        const float xv = xc[base * EDIM + e];
        const float bc = dbc[base * 48 + 16 + lane];
        const float dx = d * xv;
        float y = 0.f;
#pragma unroll
        for (int n = 0; n < NST; ++n) {
            const float bn = __shfl(bc, n, 32);
            const float cn = __shfl(bc, 16 + n, 32);
            const float dA = exp2f(d * Av[n]);
            h[n] = fmaf(dA, h[n], dx * bn);
            y = fmaf(h[n], cn, y);
        }
        y = fmaf(dsk, xv, y);
        const float z = xz[base * 1024 + 512 + e];
        xc[base * EDIM + e] = y * siluf(z);    // in-place: each elem touched once
    }
}

// --------------------------------------------------------------------------
extern "C" void kernel_launch(void* const* d_in, const int* in_sizes, int n_in,
                              void* d_out, int out_size, void* d_ws, size_t ws_size,
                              hipStream_t stream) {
    const float* x    = (const float*)d_in[0];
    const float* inW  = (const float*)d_in[1];
    const float* cw   = (const float*)d_in[2];
    const float* cb   = (const float*)d_in[3];
    const float* xpW  = (const float*)d_in[4];
    const float* dtW  = (const float*)d_in[5];
    const float* dtB  = (const float*)d_in[6];
    const float* Alog = (const float*)d_in[7];
    const float* Dsk  = (const float*)d_in[8];
    const float* outW = (const float*)d_in[9];
    const float* nw   = (const float*)d_in[10];

    float* ws = (float*)d_ws;
    size_t off = 0;
    float* seq    = ws + off; off += (size_t)MROWS * DMODEL;
    float* normed = ws + off; off += (size_t)MROWS * DMODEL;
    float* xzb    = ws + off; off += (size_t)MROWS * 1024;
    float* xcb    = ws + off; off += (size_t)MROWS * EDIM;
    float* deltab = ws + off; off += (size_t)MROWS * EDIM;
    float* dbcb   = ws + off; off += (size_t)MROWS * 48;
    float* Pb     = ws + off; off += (size_t)BB * EDIM * NST * NCH;
    float* Sb     = ws + off; off += (size_t)BB * EDIM * NST * NCH;
    float* Hb     = ws + off; off += (size_t)BB * EDIM * NST * NCH;

    transpose_in_k<<<dim3(LSEQ / 32, DMODEL / 32, BB), dim3(32, 8), 0, stream>>>(x, seq);

    for (int i = 0; i < 2; ++i) {
        const float* inWi  = inW  + (size_t)i * 1024 * DMODEL;
        const float* cwi   = cw   + (size_t)i * EDIM * 4;
        const float* cbi   = cb   + (size_t)i * EDIM;
        const float* xpWi  = xpW  + (size_t)i * 48 * EDIM;
        const float* dtWi  = dtW  + (size_t)i * EDIM * RNK;
        const float* dtBi  = dtB  + (size_t)i * EDIM;
        const float* Alogi = Alog + (size_t)i * EDIM * NST;
        const float* Dski  = Dsk  + (size_t)i * EDIM;
        const float* outWi = outW + (size_t)i * DMODEL * EDIM;
        const float* nwi   = nw   + (size_t)i * DMODEL;

        rmsnorm_k<<<MROWS / 8, 256, 0, stream>>>(seq, nwi, normed);
        // in_proj: (16384x256) x (1024x256)^T -> xz.  32x64 tiles: 8192 waves.
        gemm_wmma_f32<2, 4, 0><<<2048, 128, 0, stream>>>(
            normed, inWi, nullptr, xzb, MROWS, DMODEL, DMODEL, DMODEL, 1024);
        conv_silu_k<<<(size_t)MROWS * EDIM / 256, 256, 0, stream>>>(xzb, cwi, cbi, xcb);
        // x_proj: (16384x512) x (48x512)^T -> dbc.  32x48 tiles: 512 waves.
        gemm_wmma_f32<2, 3, 0><<<128, 128, 0, stream>>>(
            xcb, xpWi, nullptr, dbcb, MROWS, EDIM, EDIM, EDIM, 48);
        // dt_proj + bias + softplus: (16384x16 of dbc) x (512x16)^T -> delta
        gemm_wmma_f32<2, 4, 1><<<1024, 128, 0, stream>>>(
            dbcb, dtWi, dtBi, deltab, MROWS, RNK, 48, RNK, EDIM);
        // chunked selective scan
        scan_passA<<<256, 256, 0, stream>>>(deltab, xcb, dbcb, Alogi, Pb, Sb);
        scan_passB<<<BB * EDIM * NST / 256, 256, 0, stream>>>(Pb, Sb, Hb);
        scan_passC<<<256, 256, 0, stream>>>(deltab, xcb, dbcb, Alogi, Hb, Dski, xzb);
        // out_proj + residual: (16384x512) x (256x512)^T added into seq
        gemm_wmma_f32<2, 4, 2><<<512, 128, 0, stream>>>(
            xcb, outWi, seq, seq, MROWS, EDIM, EDIM, EDIM, DMODEL);
    }

    transpose_out_k<<<dim3(LSEQ / 32, DMODEL / 32, BB), dim3(32, 8), 0, stream>>>(
        seq, (float*)d_out);
}